// TGFormer_4758823764092
// MI455X (gfx1250) — compile-verified
//
#include <hip/hip_runtime.h>
#include <hip/hip_bf16.h>
#include <math.h>

// Problem constants (match reference)
#define NODES   20000
#define EDGES   320000

typedef float v2f __attribute__((ext_vector_type(2)));
typedef float v8f __attribute__((ext_vector_type(8)));

// ---- order-preserving f32 <-> u32 encoding for atomicMax on floats ----
__device__ __forceinline__ unsigned f32_ord(float f) {
  unsigned u = __float_as_uint(f);
  return (u & 0x80000000u) ? ~u : (u | 0x80000000u);
}
__device__ __forceinline__ float ord_f32(unsigned u) {
  return (u & 0x80000000u) ? __uint_as_float(u & 0x7fffffffu)
                           : __uint_as_float(~u);
}
#define ORD_NEG_INF 0x007fffffu  // f32_ord(-inf)

// ---------------------------------------------------------------------------
// C[M,N] = op(A[M,K] @ W[K,N] + bias[N]); op = relu if relu!=0.
// Requirements: M%16==0, K%4==0 and K<=256, N%64==0.
// Block = 256 threads = 8 waves; wave w computes a 16x64 tile.
// W panel (K x 64) staged in LDS in K-PAIR-INTERLEAVED layout:
//   element (k, n) at dword ((k>>1)*64 + n)*2 + (k&1)
// so each WMMA B fragment (rows k, k+1 at column n) is one ds_load_b64
// landing directly in an even VGPR pair — no packing movs.
// Uses V_WMMA_F32_16X16X4_F32 (CDNA5 f32 matrix pipe).
// ---------------------------------------------------------------------------
__launch_bounds__(256)
__global__ void gemm_bias_wmma(const float* __restrict__ A,
                               const float* __restrict__ W,
                               const float* __restrict__ bias,
                               float* __restrict__ C,
                               int M, int K, int N, int relu)
{
  __shared__ float sW[256 * 64];  // 64 KB max (K<=256)
  const int n0 = blockIdx.y * 64;

  // cooperative load of weight panel [K x 64], interleaved by K-pairs
  for (int idx = threadIdx.x; idx < K * 64; idx += 256) {
    const int k = idx >> 6;
    const int n = idx & 63;
    sW[(((k >> 1) * 64 + n) << 1) + (k & 1)] = W[(size_t)k * N + n0 + n];
  }
  __syncthreads();

  const int wave = threadIdx.x >> 5;
  const int lane = threadIdx.x & 31;
  const int mt   = blockIdx.x * 8 + wave;
  if (mt * 16 >= M) return;   // wave-uniform: EXEC stays all-ones below

  const int kh = lane >> 4;   // K half-select (0: K=k0,k0+1 ; 1: K=k0+2,k0+3)
  const int nl = lane & 15;   // N (and A-row) lane index

  const float* Arow = A + (size_t)(mt * 16 + nl) * K;
  const v2f* sW2 = (const v2f*)sW;   // pair view: [K/2][64] of (k_even,k_odd)

  v8f acc0 = {}, acc1 = {}, acc2 = {}, acc3 = {};

  for (int k0 = 0; k0 < K; k0 += 4) {
    // A fragment: 16x4 f32 striped per ISA layout (one b64 global load)
    v2f a = *(const v2f*)(Arow + k0 + 2 * kh);

    // B fragments: one b64 LDS load each
    const int pr = ((k0 >> 1) + kh) * 64 + nl;  // pair-row base for this lane
    v2f b0 = sW2[pr];
    v2f b1 = sW2[pr + 16];
    v2f b2 = sW2[pr + 32];
    v2f b3 = sW2[pr + 48];

    acc0 = __builtin_amdgcn_wmma_f32_16x16x4_f32(false, a, false, b0, (short)0, acc0, false, false);
    acc1 = __builtin_amdgcn_wmma_f32_16x16x4_f32(false, a, false, b1, (short)0, acc1, false, false);
    acc2 = __builtin_amdgcn_wmma_f32_16x16x4_f32(false, a, false, b2, (short)0, acc2, false, false);
    acc3 = __builtin_amdgcn_wmma_f32_16x16x4_f32(false, a, false, b3, (short)0, acc3, false, false);
  }

  // store: C/D layout -> lane row = kh*8 + vgpr, col = n0 + j*16 + nl
  const float bz0 = bias[n0 + nl];
  const float bz1 = bias[n0 + 16 + nl];
  const float bz2 = bias[n0 + 32 + nl];
  const float bz3 = bias[n0 + 48 + nl];
#pragma unroll
  for (int i = 0; i < 8; ++i) {
    size_t r = (size_t)(mt * 16 + kh * 8 + i) * N + n0 + nl;
    float o0 = acc0[i] + bz0;
    float o1 = acc1[i] + bz1;
    float o2 = acc2[i] + bz2;
    float o3 = acc3[i] + bz3;
    if (relu) {
      o0 = fmaxf(o0, 0.f); o1 = fmaxf(o1, 0.f);
      o2 = fmaxf(o2, 0.f); o3 = fmaxf(o3, 0.f);
    }
    C[r]      = o0;
    C[r + 16] = o1;
    C[r + 32] = o2;
    C[r + 48] = o3;
  }
}

// ---------------------------------------------------------------------------
// Edge phase kernels. heads encoded as shift (2 -> 4 heads, 0 -> 1 head);
// per-head channel count is 64 for both conv layers (scale = 1/8).
// ---------------------------------------------------------------------------
__global__ void fill_u32(unsigned* __restrict__ p, unsigned v, int n) {
  int i = blockIdx.x * blockDim.x + threadIdx.x;
  if (i < n) p[i] = v;
}

__global__ void relu_copy(const float* __restrict__ in, float* __restrict__ out, int n) {
  int i = blockIdx.x * blockDim.x + threadIdx.x;
  if (i < n) out[i] = fmaxf(in[i], 0.f);
}

// alpha[e,h] = (q[dst[e],h,:] . k[src[e],h,:]) / 8 ; segment-max into mmax[dst,h]
__global__ void edge_logits_max(const float* __restrict__ q,
                                const float* __restrict__ k,
                                const int* __restrict__ src,
                                const int* __restrict__ dst,
                                float* __restrict__ alpha,
                                unsigned* __restrict__ mmax,
                                int nE, int hshift)
{
  int t = blockIdx.x * blockDim.x + threadIdx.x;
  if (t >= (nE << hshift)) return;
  const int h = t & ((1 << hshift) - 1);
  const int e = t >> hshift;
  const int C = 64 << hshift;
  const int d = dst[e], s = src[e];
  const float4* qv = (const float4*)(q + (size_t)d * C + h * 64);
  const float4* kv = (const float4*)(k + (size_t)s * C + h * 64);
  float acc = 0.f;
#pragma unroll
  for (int i = 0; i < 16; ++i) {
    float4 a4 = qv[i], b4 = kv[i];
    acc += a4.x * b4.x + a4.y * b4.y + a4.z * b4.z + a4.w * b4.w;
  }
  acc *= 0.125f;  // 1/sqrt(64)
  alpha[t] = acc;
  atomicMax(&mmax[(size_t)(d << hshift) + h], f32_ord(acc));
}

// alpha[e,h] = exp(alpha - m[dst,h]) ; segment-sum into denom[dst,h]
__global__ void edge_exp_sum(float* __restrict__ alpha,
                             const unsigned* __restrict__ mmax,
                             float* __restrict__ denom,
                             const int* __restrict__ dst,
                             int nE, int hshift)
{
  int t = blockIdx.x * blockDim.x + threadIdx.x;
  if (t >= (nE << hshift)) return;
  const int h = t & ((1 << hshift) - 1);
  const int e = t >> hshift;
  const int idx = (dst[e] << hshift) + h;
  float m = ord_f32(mmax[idx]);
  if (!__builtin_isfinite(m)) m = 0.f;  // matches reference where(isfinite(m), m, 0)
  float ex = expf(alpha[t] - m);
  alpha[t] = ex;
  atomicAdd(&denom[idx], ex);
}

// agg[dst,h,:] += (ex / (denom+1e-16)) * v[src,h,:]
__global__ void edge_scatter_v(const float* __restrict__ alpha,
                               const float* __restrict__ denom,
                               const float* __restrict__ v,
                               const int* __restrict__ src,
                               const int* __restrict__ dst,
                               float* __restrict__ agg,
                               int nE, int hshift)
{
  int t = blockIdx.x * blockDim.x + threadIdx.x;
  if (t >= (nE << hshift)) return;
  const int h = t & ((1 << hshift) - 1);
  const int e = t >> hshift;
  const int C = 64 << hshift;
  const int d = dst[e], s = src[e];
  const float a = alpha[t] / (denom[(d << hshift) + h] + 1e-16f);
  const float4* vv = (const float4*)(v + (size_t)s * C + h * 64);
  float* out = agg + (size_t)d * C + h * 64;
#pragma unroll
  for (int i = 0; i < 16; ++i) {
    float4 x4 = vv[i];
    atomicAdd(out + i * 4 + 0, a * x4.x);
    atomicAdd(out + i * 4 + 1, a * x4.y);
    atomicAdd(out + i * 4 + 2, a * x4.z);
    atomicAdd(out + i * 4 + 3, a * x4.w);
  }
}

// ---------------------------------------------------------------------------
extern "C" void kernel_launch(void* const* d_in, const int* in_sizes, int n_in,
                              void* d_out, int out_size, void* d_ws, size_t ws_size,
                              hipStream_t stream)
{
  const float* x   = (const float*)d_in[0];
  const int*   ei  = (const int*)  d_in[1];  // [2,E] int32 (JAX default x64 off)
  const float* W1  = (const float*)d_in[2];  const float* b1  = (const float*)d_in[3];
  const float* Wq1 = (const float*)d_in[4];  const float* bq1 = (const float*)d_in[5];
  const float* Wk1 = (const float*)d_in[6];  const float* bk1 = (const float*)d_in[7];
  const float* Wv1 = (const float*)d_in[8];  const float* bv1 = (const float*)d_in[9];
  const float* Ws1 = (const float*)d_in[10]; const float* bs1 = (const float*)d_in[11];
  const float* Wq2 = (const float*)d_in[12]; const float* bq2 = (const float*)d_in[13];
  const float* Wk2 = (const float*)d_in[14]; const float* bk2 = (const float*)d_in[15];
  const float* Wv2 = (const float*)d_in[16]; const float* bv2 = (const float*)d_in[17];
  const float* Ws2 = (const float*)d_in[18]; const float* bs2 = (const float*)d_in[19];
  (void)in_sizes; (void)n_in; (void)out_size; (void)ws_size;

  const int* src = ei;          // edge_index[0]
  const int* dst = ei + EDGES;  // edge_index[1]

  // Workspace layout (floats); peak 23.2M floats = 92.8 MB, with reuse.
  float* ws    = (float*)d_ws;
  float* h0    = ws;                 // N*64
  float* q1    = h0   + 1280000;     // N*256  (reused as h1)
  float* k1    = q1   + 5120000;     // N*256  (reused as q2,k2,v2)
  float* v1    = k1   + 5120000;     // N*256
  float* agg1  = v1   + 5120000;     // N*256  (skip-initialized)
  float* alpha = agg1 + 5120000;     // E*4    (reused for conv2)
  unsigned* mx = (unsigned*)(alpha + 1280000);  // N*4 u32
  float* dn    = (float*)(mx + 80000);          // N*4

  const dim3 blk(256);
  const dim3 gN64(157, 1);   // ceil(1250 m-tiles / 8 waves)
  const dim3 gN256(157, 4);

  // ---- layer 0: h0 = relu(x @ W1 + b1) ----
  gemm_bias_wmma<<<gN64, blk, 0, stream>>>(x, W1, b1, h0, NODES, 128, 64, 1);

  // ---- conv1 projections (agg1 pre-loaded with skip h0@Ws1+bs1) ----
  gemm_bias_wmma<<<gN256, blk, 0, stream>>>(h0, Wq1, bq1, q1,   NODES, 64, 256, 0);
  gemm_bias_wmma<<<gN256, blk, 0, stream>>>(h0, Wk1, bk1, k1,   NODES, 64, 256, 0);
  gemm_bias_wmma<<<gN256, blk, 0, stream>>>(h0, Wv1, bv1, v1,   NODES, 64, 256, 0);
  gemm_bias_wmma<<<gN256, blk, 0, stream>>>(h0, Ws1, bs1, agg1, NODES, 64, 256, 0);

  // ---- conv1 edge softmax + aggregate (4 heads) ----
  fill_u32<<<(80000 + 255) / 256, blk, 0, stream>>>(mx, ORD_NEG_INF, 80000);
  fill_u32<<<(80000 + 255) / 256, blk, 0, stream>>>((unsigned*)dn, 0u, 80000);
  const int T1 = EDGES * 4;
  edge_logits_max<<<(T1 + 255) / 256, blk, 0, stream>>>(q1, k1, src, dst, alpha, mx, EDGES, 2);
  edge_exp_sum  <<<(T1 + 255) / 256, blk, 0, stream>>>(alpha, mx, dn, dst, EDGES, 2);
  edge_scatter_v<<<(T1 + 255) / 256, blk, 0, stream>>>(alpha, dn, v1, src, dst, agg1, EDGES, 2);

  // ---- h1 = relu(conv1 out) ----
  float* h1 = q1;  // reuse
  relu_copy<<<(5120000 + 255) / 256, blk, 0, stream>>>(agg1, h1, 5120000);

  // ---- conv2 projections (agg2 = d_out pre-loaded with skip h1@Ws2+bs2) ----
  float* q2 = k1;             // reuse k1 region
  float* k2 = k1 + 1280000;
  float* v2 = k1 + 2560000;
  float* out = (float*)d_out; // N*64 — heads=1, concat=False -> mean == identity
  gemm_bias_wmma<<<gN64, blk, 0, stream>>>(h1, Wq2, bq2, q2,  NODES, 256, 64, 0);
  gemm_bias_wmma<<<gN64, blk, 0, stream>>>(h1, Wk2, bk2, k2,  NODES, 256, 64, 0);
  gemm_bias_wmma<<<gN64, blk, 0, stream>>>(h1, Wv2, bv2, v2,  NODES, 256, 64, 0);
  gemm_bias_wmma<<<gN64, blk, 0, stream>>>(h1, Ws2, bs2, out, NODES, 256, 64, 0);

  // ---- conv2 edge softmax + aggregate (1 head) ----
  fill_u32<<<(20000 + 255) / 256, blk, 0, stream>>>(mx, ORD_NEG_INF, 20000);
  fill_u32<<<(20000 + 255) / 256, blk, 0, stream>>>((unsigned*)dn, 0u, 20000);
  edge_logits_max<<<(EDGES + 255) / 256, blk, 0, stream>>>(q2, k2, src, dst, alpha, mx, EDGES, 0);
  edge_exp_sum  <<<(EDGES + 255) / 256, blk, 0, stream>>>(alpha, mx, dn, dst, EDGES, 0);
  edge_scatter_v<<<(EDGES + 255) / 256, blk, 0, stream>>>(alpha, dn, v2, src, dst, out, EDGES, 0);
}